// GCNExplainerWrapper_43989055045752
// MI455X (gfx1250) — compile-verified
//
#include <hip/hip_runtime.h>
#include <hip/hip_bf16.h>

typedef __attribute__((ext_vector_type(2))) float v2f;
typedef __attribute__((ext_vector_type(8))) float v8f;

static __host__ __device__ inline int cdiv(int a, int b) { return (a + b - 1) / b; }

#define MTILES 8   // M-tiles (of 16 rows) processed per block in the WMMA GEMM

// ---------------------------------------------------------------------------
// Degree / normalization kernels
// ---------------------------------------------------------------------------
__global__ void k_init_one(float* __restrict__ deg, int n) {
    int i = blockIdx.x * blockDim.x + threadIdx.x;
    if (i < n) deg[i] = 1.0f;   // self-loop contributes 1 to every node's degree
}

__global__ void k_count_deg(const int* __restrict__ dst, float* __restrict__ deg, int e) {
    int i = blockIdx.x * blockDim.x + threadIdx.x;
    if (i < e) atomicAdd(&deg[dst[i]], 1.0f);
}

__global__ void k_rsqrt(float* __restrict__ deg, int n) {
    int i = blockIdx.x * blockDim.x + threadIdx.x;
    if (i < n) deg[i] = rsqrtf(deg[i]);   // deg >= 1 always (self-loops)
}

// ---------------------------------------------------------------------------
// WMMA fp32 GEMM: Y[M x Nout] = X[M x 64] @ W[64 x Nout]  (+bias, +relu)
// K fixed at 64. One block = MTILES consecutive 16-row M tiles; wave w owns
// output columns [16w, 16w+16). B fragments (16 x v2f = 32 VGPRs) are loaded
// ONCE per wave and reused across all M tiles -> 8x less weight traffic,
// 128 v_wmma per wave per block. Uses V_WMMA_F32_16X16X4_F32 (exact fp32).
//
// ISA layouts (wave32, 32-bit elements), half = lane>>4, l16 = lane&15:
//   A (16x4): lane row M=l16, VGPR v <-> K = v + 2*half
//   B (4x16): lane col N=l16, VGPR v <-> K = v + 2*half
//   C/D (16x16): lane col N=l16, VGPR v <-> M = v + 8*half
// ---------------------------------------------------------------------------
__global__ void k_gemm_wmma(const float* __restrict__ X,
                            const float* __restrict__ W,
                            const float* __restrict__ bias,   // nullptr = none
                            float* __restrict__ Y,
                            int M, int Nout, int do_relu) {
    const int lane = threadIdx.x & 31;
    const int wave = threadIdx.x >> 5;
    const int half = lane >> 4;
    const int l16  = lane & 15;
    const int n0   = wave * 16;
    const int n    = n0 + l16;

    // --- preload all B fragments for K=64 (16 k-steps), kept in registers
    v2f bfrag[16];
#pragma unroll
    for (int s = 0; s < 16; ++s) {
        const int ka = 4 * s + 2 * half;
        bfrag[s].x = W[(size_t)ka       * Nout + n];
        bfrag[s].y = W[(size_t)(ka + 1) * Nout + n];
    }
    const float bv = bias ? bias[n] : 0.0f;

    int m0 = blockIdx.x * (16 * MTILES);
#pragma unroll
    for (int t = 0; t < MTILES; ++t, m0 += 16) {
        if (m0 >= M) break;                                    // wave-uniform
        const int row = (m0 + l16 < M) ? (m0 + l16) : (M - 1); // clamp, EXEC unchanged
        const float* __restrict__ xrow = X + (size_t)row * 64;

        v8f acc = {};
#pragma unroll
        for (int s = 0; s < 16; ++s) {
            const int ka = 4 * s + 2 * half;
            v2f a;
            a.x = xrow[ka];
            a.y = xrow[ka + 1];
            acc = __builtin_amdgcn_wmma_f32_16x16x4_f32(
                /*neg_a=*/false, a, /*neg_b=*/false, bfrag[s],
                /*c_mod=*/(short)0, acc, /*reuse_a=*/false, /*reuse_b=*/false);
        }

#pragma unroll
        for (int v = 0; v < 8; ++v) {
            const int m = m0 + v + 8 * half;
            if (m < M) {
                float val = acc[v] + bv;
                if (do_relu) val = fmaxf(val, 0.0f);
                Y[(size_t)m * Nout + n] = val;
            }
        }
    }
}

// ---------------------------------------------------------------------------
// Message passing (feature width fixed at 64 = 16 float4)
// ---------------------------------------------------------------------------
// agg[i,:] = t[i,:] * dinv[i]^2   (self-loop term; also initializes agg)
__global__ void k_selfloop_init(const float4* __restrict__ t,
                                const float* __restrict__ dinv,
                                float4* __restrict__ agg, int total4) {
    int idx = blockIdx.x * blockDim.x + threadIdx.x;
    if (idx >= total4) return;
    const float w  = dinv[idx >> 4];
    const float w2 = w * w;
    float4 v = t[idx];
    v.x *= w2; v.y *= w2; v.z *= w2; v.w *= w2;
    agg[idx] = v;
}

// agg[dst[e], :] += t[src[e], :] * dinv[src]*dinv[dst]
// 16 threads per edge, float4 gathers (b128), 4 f32 atomics each (L2-resident).
__global__ void k_scatter_edges(const int* __restrict__ src,
                                const int* __restrict__ dst,
                                const float* __restrict__ dinv,
                                const float4* __restrict__ t,
                                float* __restrict__ agg, int total16) {
    int idx = blockIdx.x * blockDim.x + threadIdx.x;
    if (idx >= total16) return;
    const int e  = idx >> 4;
    const int f4 = idx & 15;
    const int s = src[e];
    const int d = dst[e];
    const float w = dinv[s] * dinv[d];
    float4 v = t[(size_t)s * 16 + f4];
    float* p = agg + (size_t)d * 64 + f4 * 4;
    atomicAdd(p + 0, v.x * w);
    atomicAdd(p + 1, v.y * w);
    atomicAdd(p + 2, v.z * w);
    atomicAdd(p + 3, v.w * w);
}

// in-place h = relu(h + bias[f]) over N*64 elements (float4-wide)
__global__ void k_bias_relu(float4* __restrict__ h, const float4* __restrict__ b, int total4) {
    int idx = blockIdx.x * blockDim.x + threadIdx.x;
    if (idx >= total4) return;
    const float4 bb = b[idx & 15];
    float4 v = h[idx];
    v.x = fmaxf(v.x + bb.x, 0.0f);
    v.y = fmaxf(v.y + bb.y, 0.0f);
    v.z = fmaxf(v.z + bb.z, 0.0f);
    v.w = fmaxf(v.w + bb.w, 0.0f);
    h[idx] = v;
}

// ---------------------------------------------------------------------------
// Decoder tail: pred[i] = dot(h3[i, 0:32], Wd2) + bd2
// ---------------------------------------------------------------------------
__global__ void k_decoder_out(const float4* __restrict__ h3,
                              const float4* __restrict__ Wd2,
                              const float* __restrict__ bd2,
                              float* __restrict__ out, int n) {
    int i = blockIdx.x * blockDim.x + threadIdx.x;
    if (i >= n) return;
    float s = bd2[0];
#pragma unroll
    for (int f = 0; f < 8; ++f) {
        const float4 a = h3[(size_t)i * 8 + f];
        const float4 w = Wd2[f];
        s = fmaf(a.x, w.x, s);
        s = fmaf(a.y, w.y, s);
        s = fmaf(a.z, w.z, s);
        s = fmaf(a.w, w.w, s);
    }
    out[i] = s;
}

// ---------------------------------------------------------------------------
extern "C" void kernel_launch(void* const* d_in, const int* in_sizes, int n_in,
                              void* d_out, int out_size, void* d_ws, size_t ws_size,
                              hipStream_t stream) {
    const float* x   = (const float*)d_in[0];
    const int*   ei  = (const int*)  d_in[1];   // (2, E) flat: row0 = src, row1 = dst
    const float* W1  = (const float*)d_in[2];
    const float* b1  = (const float*)d_in[3];
    const float* W2  = (const float*)d_in[4];
    const float* b2  = (const float*)d_in[5];
    const float* Wd1 = (const float*)d_in[6];
    const float* bd1 = (const float*)d_in[7];
    const float* Wd2 = (const float*)d_in[8];
    const float* bd2 = (const float*)d_in[9];
    float* out = (float*)d_out;

    const int N = in_sizes[0] / 64;      // 50000
    const int E = in_sizes[1] / 2;       // 800000
    const int* src = ei;
    const int* dst = ei + E;

    float* ws   = (float*)d_ws;
    float* dinv = ws;                            // N floats
    float* bufA = dinv + N;                      // N*64 floats (transform out)
    float* bufB = bufA + (size_t)N * 64;         // N*64 floats (aggregation / h)

    const int T = 256;
    const int nf4    = N * 16;                   // node-feature float4 elements
    const int ef16   = E * 16;                   // edge-scatter work items (float4 granules)
    const int gblk   = cdiv(N, 16 * MTILES);     // GEMM blocks

    // --- normalization: dinv = (1 + in-degree)^(-1/2)
    k_init_one <<<cdiv(N, T), T, 0, stream>>>(dinv, N);
    k_count_deg<<<cdiv(E, T), T, 0, stream>>>(dst, dinv, E);
    k_rsqrt    <<<cdiv(N, T), T, 0, stream>>>(dinv, N);

    // --- layer 1: bufA = x@W1 ; bufB = A_hat * bufA ; h1 = relu(bufB + b1)
    k_gemm_wmma<<<gblk, 32 * (64 / 16), 0, stream>>>(x, W1, nullptr, bufA, N, 64, 0);
    k_selfloop_init<<<cdiv(nf4, T), T, 0, stream>>>((const float4*)bufA, dinv, (float4*)bufB, nf4);
    k_scatter_edges<<<cdiv(ef16, T), T, 0, stream>>>(src, dst, dinv, (const float4*)bufA, bufB, ef16);
    k_bias_relu<<<cdiv(nf4, T), T, 0, stream>>>((float4*)bufB, (const float4*)b1, nf4);

    // --- layer 2: bufA = h1@W2 ; bufB = A_hat * bufA ; h2 = relu(bufB + b2)
    k_gemm_wmma<<<gblk, 32 * (64 / 16), 0, stream>>>(bufB, W2, nullptr, bufA, N, 64, 0);
    k_selfloop_init<<<cdiv(nf4, T), T, 0, stream>>>((const float4*)bufA, dinv, (float4*)bufB, nf4);
    k_scatter_edges<<<cdiv(ef16, T), T, 0, stream>>>(src, dst, dinv, (const float4*)bufA, bufB, ef16);
    k_bias_relu<<<cdiv(nf4, T), T, 0, stream>>>((float4*)bufB, (const float4*)b2, nf4);

    // --- decoder: h3 = relu(h2 @ Wd1 + bd1)   (N x 32, stored in bufA)
    k_gemm_wmma<<<gblk, 32 * (32 / 16), 0, stream>>>(bufB, Wd1, bd1, bufA, N, 32, 1);

    // --- pred = h3 @ Wd2 + bd2
    k_decoder_out<<<cdiv(N, T), T, 0, stream>>>((const float4*)bufA, (const float4*)Wd2, bd2, out, N);
}